// DecoderrKBMemNN_27917287424596
// MI455X (gfx1250) — compile-verified
//
#include <hip/hip_runtime.h>
#include <math.h>

// Problem constants (match reference)
#define VOCAB   32000
#define EDIM    128
#define NB      16
#define NM      4096
#define NS      4
#define CHUNKS  8            // split of M for the weighted-sum pass

typedef __attribute__((ext_vector_type(2))) float v2f;
typedef __attribute__((ext_vector_type(8))) float v8f;

// -------------------------------------------------------------------------
// Kernel 1: logits[b,m] = dot( sum_s C[idx_s], u[b] )  via V_WMMA_F32_16X16X4_F32
// grid = NB * (NM/64) blocks, 128 threads (4 waves, 16 m-rows per wave)
// -------------------------------------------------------------------------
__global__ __launch_bounds__(128) void logits_wmma_kernel(
    const int* __restrict__ story,      // [NB, NM, NS]
    const float* __restrict__ Ctab,     // one table [VOCAB, EDIM]
    const float* __restrict__ u,        // [NB, EDIM]
    float* __restrict__ out)            // [NB, NM]
{
    __shared__ float ms[64][EDIM + 4];  // +4 float pad -> conflict-free ds_load_b64 in WMMA feed
    __shared__ float su[EDIM];
    __shared__ int   sidx[64 * NS];

    const int tilesPerB = NM / 64;
    const int b     = blockIdx.x / tilesPerB;
    const int mBase = (blockIdx.x % tilesPerB) * 64;
    const int t     = threadIdx.x;

    // stage u[b] and the 256 token indices for this tile
    su[t] = u[b * EDIM + t];
    {
        const int base = (b * NM + mBase) * NS;
        sidx[t]       = story[base + t];
        sidx[t + 128] = story[base + 128 + t];
    }
    __syncthreads();

    // Phase 1: coalesced gather-and-sum. thread t owns column d = t.
    // Each table-row read is a full 512B coalesced line across the block.
    for (int ml = 0; ml < 64; ++ml) {
        const int i0 = sidx[ml * NS + 0];
        const int i1 = sidx[ml * NS + 1];
        const int i2 = sidx[ml * NS + 2];
        const int i3 = sidx[ml * NS + 3];
        ms[ml][t] = Ctab[(size_t)i0 * EDIM + t] + Ctab[(size_t)i1 * EDIM + t] +
                    Ctab[(size_t)i2 * EDIM + t] + Ctab[(size_t)i3 * EDIM + t];
    }
    __syncthreads();

    // Phase 2: WMMA dot products. Wave w handles m rows [16w, 16w+16).
    // A (16x4 f32): VGPR0 = K {0|2}, VGPR1 = K {1|3}; lanes 0-15 -> K 0/1, lanes 16-31 -> K 2/3.
    // B (4x16): column j = u chunk for all j (replicated broadcast from LDS).
    const int w    = t >> 5;
    const int lane = t & 31;
    const int rowA = lane & 15;
    const int kOff = (lane < 16) ? 0 : 2;
    const float* arow = &ms[w * 16 + rowA][0];

    v8f c = {0.f, 0.f, 0.f, 0.f, 0.f, 0.f, 0.f, 0.f};
    for (int k0 = 0; k0 < EDIM; k0 += 4) {
        v2f a, bb;
        a.x  = arow[k0 + kOff];
        a.y  = arow[k0 + kOff + 1];
        bb.x = su[k0 + kOff];
        bb.y = su[k0 + kOff + 1];
        c = __builtin_amdgcn_wmma_f32_16x16x4_f32(false, a, false, bb,
                                                  (short)0, c, false, false);
    }

    // C/D layout: lane 0 VGPR r -> (M=r, N=0); lane 16 VGPR r -> (M=8+r, N=0).
    if (lane == 0 || lane == 16) {
        const int mo = mBase + w * 16 + ((lane == 0) ? 0 : 8);
        float* dst = out + (size_t)b * NM + mo;
        float4 lo = make_float4(c[0], c[1], c[2], c[3]);
        float4 hi = make_float4(c[4], c[5], c[6], c[7]);
        *(float4*)(dst)     = lo;
        *(float4*)(dst + 4) = hi;
    }
}

// -------------------------------------------------------------------------
// Kernel 2: softmax stats (max, sumexp) per batch row. grid = NB, 256 threads.
// -------------------------------------------------------------------------
__global__ __launch_bounds__(256) void stats_kernel(
    const float* __restrict__ logits,   // [NB, NM]
    float* __restrict__ stats)          // [NB, 2] = {max, sumexp}
{
    __shared__ float red[256];
    const int b = blockIdx.x, t = threadIdx.x;

    float mx = -1e30f;
    for (int i = t; i < NM; i += 256) mx = fmaxf(mx, logits[(size_t)b * NM + i]);
    red[t] = mx; __syncthreads();
    for (int s = 128; s > 0; s >>= 1) {
        if (t < s) red[t] = fmaxf(red[t], red[t + s]);
        __syncthreads();
    }
    const float smax = red[0];
    __syncthreads();

    float se = 0.f;
    for (int i = t; i < NM; i += 256) se += __expf(logits[(size_t)b * NM + i] - smax);
    red[t] = se; __syncthreads();
    for (int s = 128; s > 0; s >>= 1) {
        if (t < s) red[t] += red[t + s];
        __syncthreads();
    }
    if (t == 0) { stats[b * 2] = smax; stats[b * 2 + 1] = red[0]; }
}

// -------------------------------------------------------------------------
// Kernel 3: partial weighted sum  o_part[b,chunk,d] = sum_{m in chunk} p[m] * msum_{h+1}[b,m,d]
// grid = NB*CHUNKS blocks, 256 threads (8 waves; each lane owns 4 d's -> float4 row loads)
// -------------------------------------------------------------------------
__global__ __launch_bounds__(256) void attend_partial_kernel(
    const int* __restrict__ story,
    const float* __restrict__ Ctab,     // table for hop+1
    const float* __restrict__ logits,
    const float* __restrict__ stats,
    float* __restrict__ partial)        // [NB*CHUNKS, EDIM]
{
    __shared__ float osum[8][EDIM];
    const int blk = blockIdx.x;
    const int b   = blk / CHUNKS;
    const int ch  = blk % CHUNKS;
    const int t    = threadIdx.x;
    const int w    = t >> 5;
    const int lane = t & 31;

    const float smax = stats[b * 2];
    const float inv  = 1.0f / stats[b * 2 + 1];

    float a0 = 0.f, a1 = 0.f, a2 = 0.f, a3 = 0.f;
    const int mStart = ch * (NM / CHUNKS);
    const int mEnd   = mStart + (NM / CHUNKS);

    for (int m = mStart + w; m < mEnd; m += 8) {
        const int* st = story + (size_t)(b * NM + m) * NS;
        const int i0 = st[0], i1 = st[1], i2 = st[2], i3 = st[3];
        const float p = __expf(logits[(size_t)b * NM + m] - smax) * inv;
        // each lane reads float4 at d = lane*4 -> 512B coalesced per row
        const float4 v0 = *(const float4*)(Ctab + (size_t)i0 * EDIM + lane * 4);
        const float4 v1 = *(const float4*)(Ctab + (size_t)i1 * EDIM + lane * 4);
        const float4 v2 = *(const float4*)(Ctab + (size_t)i2 * EDIM + lane * 4);
        const float4 v3 = *(const float4*)(Ctab + (size_t)i3 * EDIM + lane * 4);
        a0 += p * (v0.x + v1.x + v2.x + v3.x);
        a1 += p * (v0.y + v1.y + v2.y + v3.y);
        a2 += p * (v0.z + v1.z + v2.z + v3.z);
        a3 += p * (v0.w + v1.w + v2.w + v3.w);
    }
    osum[w][lane * 4 + 0] = a0;
    osum[w][lane * 4 + 1] = a1;
    osum[w][lane * 4 + 2] = a2;
    osum[w][lane * 4 + 3] = a3;
    __syncthreads();

    if (t < EDIM) {
        float o = 0.f;
        for (int w2 = 0; w2 < 8; ++w2) o += osum[w2][t];
        partial[(size_t)blk * EDIM + t] = o;
    }
}

// -------------------------------------------------------------------------
// Kernel 4: u_out = u_in + sum_chunks partial. grid = NB, 128 threads.
// -------------------------------------------------------------------------
__global__ __launch_bounds__(128) void ureduce_kernel(
    const float* __restrict__ partial,
    const float* __restrict__ u_in,
    float* __restrict__ u_out,
    float* __restrict__ u_extra)        // optional second destination (u1 -> d_out)
{
    const int b = blockIdx.x, t = threadIdx.x;
    float o = 0.f;
    for (int ch = 0; ch < CHUNKS; ++ch)
        o += partial[(size_t)(b * CHUNKS + ch) * EDIM + t];
    const float un = u_in[b * EDIM + t] + o;
    u_out[b * EDIM + t] = un;
    if (u_extra) u_extra[b * EDIM + t] = un;
}

// -------------------------------------------------------------------------
extern "C" void kernel_launch(void* const* d_in, const int* in_sizes, int n_in,
                              void* d_out, int out_size, void* d_ws, size_t ws_size,
                              hipStream_t stream) {
    const int*   story = (const int*)d_in[0];          // [16,4096,4]
    const float* q     = (const float*)d_in[1];        // [16,128]
    const float* C     = (const float*)d_in[2];        // [4,32000,128]

    float* out = (float*)d_out;                        // [0,65536): prob_lg, [65536,67584): u1
    float* ws  = (float*)d_ws;

    float* wsLogits  = ws;                             // 65536 floats
    float* wsStats   = ws + 65536;                     // 32 floats
    float* wsPartial = ws + 65536 + 32;                // NB*CHUNKS*EDIM = 16384 floats
    float* wsU1      = ws + 65536 + 32 + 16384;        // 2048 floats
    float* wsU2      = wsU1 + NB * EDIM;               // 2048 floats

    const size_t tabStride = (size_t)VOCAB * EDIM;
    const dim3 gLog(NB * (NM / 64)), bLog(128);
    const dim3 gB(NB), b256(256), b128(128);
    const dim3 gPart(NB * CHUNKS);

    // ---- hop 0 : u = enc_query ----
    logits_wmma_kernel<<<gLog, bLog, 0, stream>>>(story, C + 0 * tabStride, q, wsLogits);
    stats_kernel<<<gB, b256, 0, stream>>>(wsLogits, wsStats);
    attend_partial_kernel<<<gPart, b256, 0, stream>>>(story, C + 1 * tabStride,
                                                      wsLogits, wsStats, wsPartial);
    ureduce_kernel<<<gB, b128, 0, stream>>>(wsPartial, q, wsU1, out + NB * NM); // u1 -> d_out tail

    // ---- hop 1 : u = u1 ----
    logits_wmma_kernel<<<gLog, bLog, 0, stream>>>(story, C + 1 * tabStride, wsU1, wsLogits);
    stats_kernel<<<gB, b256, 0, stream>>>(wsLogits, wsStats);
    attend_partial_kernel<<<gPart, b256, 0, stream>>>(story, C + 2 * tabStride,
                                                      wsLogits, wsStats, wsPartial);
    ureduce_kernel<<<gB, b128, 0, stream>>>(wsPartial, wsU1, wsU2, nullptr);

    // ---- hop 2 : only pre-softmax logits are returned ----
    logits_wmma_kernel<<<gLog, bLog, 0, stream>>>(story, C + 2 * tabStride, wsU2, out);
}